// YoloV3Loss_56925496541417
// MI455X (gfx1250) — compile-verified
//
#include <hip/hip_runtime.h>
#include <math.h>

#define NUM_CLASSES 80
#define NB 32          // batch
#define NANN 50        // annotations per batch
#define REC_U32 12     // record stride in u32 (48 bytes)

__constant__ float c_aw[9] = {10.f,16.f,33.f,30.f,62.f,59.f,116.f,156.f,373.f};
__constant__ float c_ah[9] = {13.f,30.f,23.f,61.f,45.f,119.f,90.f,198.f,326.f};
__constant__ int   c_dim[3] = {52,26,13};

typedef float v2f __attribute__((ext_vector_type(2)));
typedef float v8f __attribute__((ext_vector_type(8)));

// 32-lane wave sum via V_WMMA_F32_16X16X4_F32.
// A(16x4): VGPR0 = lane value (K=0 rows for lanes 0-15, K=2 for lanes 16-31),
// VGPR1 = 0  =>  A[m,0]=v[m], A[m,2]=v[m+16], others 0.
// B(4x16) = ones  =>  D[m,n] = v[m] + v[m+16].
// D vgpr r holds M=r (lanes 0-15) and M=8+r (lanes 16-31), so
// sum(c[0..7]) at lane0 = sum_{m=0..7}(v[m]+v[m+16]) and at lane16 = m=8..15.
__device__ inline float wave_reduce32(float v) {
  v2f a; a[0] = v;   a[1] = 0.0f;
  v2f b; b[0] = 1.0f; b[1] = 1.0f;
  v8f c = {};
  c = __builtin_amdgcn_wmma_f32_16x16x4_f32(false, a, false, b, (short)0, c, false, false);
  float t = c[0]+c[1]+c[2]+c[3]+c[4]+c[5]+c[6]+c[7];
  return __shfl(t, 0, 32) + __shfl(t, 16, 32);
}

// ---------------------------------------------------------------------------
// Kernel 1: per-annotation encode (order-independent part of the lax.scan).
// Each annotation maps to exactly one (layer, cell, anchor) assignment.
// ---------------------------------------------------------------------------
__global__ void yolo_encode_kernel(const float* __restrict__ target,
                                   unsigned int* __restrict__ recs) {
  int idx = blockIdx.x * blockDim.x + threadIdx.x;
  if (idx >= NB * NANN) return;
  const float* ann = target + (size_t)idx * 5;
  float cx = ann[0], cy = ann[1], w = ann[2], h = ann[3], cf = ann[4];
  bool valid = (cx + cy + w + h + cf) > 0.0f;

  float gw = w * 416.0f, gh = h * 416.0f;
  float iou[9];
  int best = 0; float bi = -1.0f;
#pragma unroll
  for (int n = 0; n < 9; ++n) {
    float aw = c_aw[n], ah = c_ah[n];
    float inter = fminf(gw, aw) * fminf(gh, ah);
    float uni   = gw * gh + aw * ah - inter;
    iou[n] = inter / (uni + 1e-16f);
    if (iou[n] > bi) { bi = iou[n]; best = n; }   // first max wins (argmax tie rule)
  }
  int layer = best / 3;
  int bl    = best - 3 * layer;
  int W     = c_dim[layer];

  float gx = cx * (float)W, gy = cy * (float)W;
  int gi = (int)gx; gi = gi < 0 ? 0 : (gi > W - 1 ? W - 1 : gi);
  int gj = (int)gy; gj = gj < 0 ? 0 : (gj > W - 1 ? W - 1 : gj);
  int cid = (int)cf; cid = cid < 0 ? 0 : (cid > NUM_CLASSES - 1 ? NUM_CLASSES - 1 : cid);

  // ignore bits: anchors of this layer whose IoU > 0.5 zero noobj at (a,gj,gi)
  unsigned ign = (iou[3*layer]   > 0.5f ? 1u : 0u)
               | (iou[3*layer+1] > 0.5f ? 2u : 0u)
               | (iou[3*layer+2] > 0.5f ? 4u : 0u);

  unsigned int* r = recs + (size_t)idx * REC_U32;
  r[0]  = valid ? (unsigned)layer : 0xFFFFFFFFu;
  r[1]  = (unsigned)gi;
  r[2]  = (unsigned)gj;
  r[3]  = (unsigned)bl;
  r[4]  = __float_as_uint(gx - (float)gi);          // tx
  r[5]  = __float_as_uint(gy - (float)gj);          // ty
  r[6]  = __float_as_uint(gw / c_aw[best]);         // tw = w*416/anchor_w
  r[7]  = __float_as_uint(gh / c_ah[best]);         // th
  r[8]  = (unsigned)cid;
  r[9]  = ign;
  r[10] = 0u; r[11] = 0u;
}

// ---------------------------------------------------------------------------
// Kernel 2: per-cell loss. Each thread = one (b, layer, anchor, j, i) cell.
// Replays the 50 records in order (from LDS) to reproduce scan overwrite
// semantics exactly. Only masked cells read the 84 box/class channels.
// ---------------------------------------------------------------------------
__global__ void yolo_loss_kernel(const float* __restrict__ p3,
                                 const float* __restrict__ p4,
                                 const float* __restrict__ p5,
                                 const unsigned int* __restrict__ recs,
                                 float* __restrict__ partials) {
  const int layer = blockIdx.z;
  const int b     = blockIdx.y;
  const int W     = c_dim[layer];
  const int HW    = W * W;
  const int ncell = 3 * HW;
  const int cell  = blockIdx.x * blockDim.x + threadIdx.x;

  __shared__ unsigned int srec[NANN * REC_U32];
  __shared__ float wsum[8];

  const unsigned int* g = recs + (size_t)b * NANN * REC_U32;
  for (int k = threadIdx.x; k < NANN * REC_U32; k += blockDim.x) srec[k] = g[k];
  __syncthreads();

  float contrib = 0.0f;
  if (cell < ncell) {
    int a   = cell / HW;
    int rem = cell - a * HW;
    int j   = rem / W;
    int i   = rem - j * W;
    const float* p  = (layer == 0) ? p3 : ((layer == 1) ? p4 : p5);
    const float* pb = p + ((size_t)b * 255 + (size_t)a * 85) * HW + rem;

    // ordered replay of annotations for this batch
    bool  mask = false, noobj = true;
    float tx = 0.f, ty = 0.f, tw = 0.f, th = 0.f;
    unsigned cls0 = 0u, cls1 = 0u, cls2 = 0u;
    for (int t = 0; t < NANN; ++t) {
      const unsigned* r = &srec[t * REC_U32];
      if ((int)r[0] == layer && (int)r[1] == i && (int)r[2] == j) {
        if ((r[9] >> a) & 1u) noobj = false;
        if ((int)r[3] == a) {
          mask = true;
          tx = __uint_as_float(r[4]); ty = __uint_as_float(r[5]);
          tw = __uint_as_float(r[6]); th = __uint_as_float(r[7]);
          int cid = (int)r[8];
          if (cid < 32)      cls0 |= 1u << cid;
          else if (cid < 64) cls1 |= 1u << (cid - 32);
          else               cls2 |= 1u << (cid - 64);
        }
      }
    }

    float conf  = pb[4 * HW];
    float sconf = 1.0f / (1.0f + expf(-conf));
    if (noobj) contrib += sconf * sconf;                 // LAMBDA_NOOBJ = 1
    if (mask) {
      float px = 1.0f / (1.0f + expf(-pb[0]));
      float py = 1.0f / (1.0f + expf(-pb[1 * HW]));
      float pw = expf(pb[2 * HW]);
      float ph = expf(pb[3 * HW]);
      float dx = px - tx, dy = py - ty, dw = pw - tw, dh = ph - th;
      contrib += dx * dx + dy * dy + dw * dw + dh * dh;  // LAMBDA_COORD = 1
      float dc = sconf - 1.0f;
      contrib += 5.0f * dc * dc;                         // LAMBDA_OBJ = 5
      for (int c = 0; c < NUM_CLASSES; ++c) {
        float pc  = 1.0f / (1.0f + expf(-pb[(5 + c) * HW]));
        float lp  = fmaxf(logf(pc),     -100.0f);
        float l1p = fmaxf(log1pf(-pc),  -100.0f);
        unsigned bit = (c < 32) ? ((cls0 >> c) & 1u)
                     : (c < 64) ? ((cls1 >> (c - 32)) & 1u)
                                : ((cls2 >> (c - 64)) & 1u);
        contrib += bit ? -lp : -l1p;                     // LAMBDA_CLASS = 1
      }
    }
  }

  // EXEC is all-1s here (no early exits) -> WMMA wave reduction is legal.
  float wtot = wave_reduce32(contrib);
  int lane = threadIdx.x & 31, wid = threadIdx.x >> 5;
  if (lane == 0) wsum[wid] = wtot;
  __syncthreads();
  if (threadIdx.x == 0) {
    float s = 0.0f;
#pragma unroll
    for (int k = 0; k < 8; ++k) s += wsum[k];
    int bid = blockIdx.x + gridDim.x * (blockIdx.y + gridDim.y * blockIdx.z);
    partials[bid] = s;
  }
}

// ---------------------------------------------------------------------------
// Kernel 3: deterministic final reduction (single wave), / B.
// ---------------------------------------------------------------------------
__global__ void yolo_finalize_kernel(const float* __restrict__ partials, int n,
                                     float* __restrict__ out) {
  float acc = 0.0f;
  for (int k = threadIdx.x; k < n; k += 32) acc += partials[k];
  float tot = wave_reduce32(acc);
  if (threadIdx.x == 0) out[0] = tot * (1.0f / (float)NB);
}

extern "C" void kernel_launch(void* const* d_in, const int* in_sizes, int n_in,
                              void* d_out, int out_size, void* d_ws, size_t ws_size,
                              hipStream_t stream) {
  const float* p3     = (const float*)d_in[0];
  const float* p4     = (const float*)d_in[1];
  const float* p5     = (const float*)d_in[2];
  const float* target = (const float*)d_in[3];

  unsigned int* recs   = (unsigned int*)d_ws;
  float* partials = (float*)((char*)d_ws + (size_t)NB * NANN * REC_U32 * sizeof(unsigned int));

  yolo_encode_kernel<<<(NB * NANN + 127) / 128, 128, 0, stream>>>(target, recs);

  // grid.x = 32 covers the largest layer: 3*52*52 = 8112 cells / 256 = 31.7
  dim3 grid(32, NB, 3);
  yolo_loss_kernel<<<grid, 256, 0, stream>>>(p3, p4, p5, recs, partials);

  int nparts = 32 * NB * 3;  // every block writes its slot (0 if out of range)
  yolo_finalize_kernel<<<1, 32, 0, stream>>>(partials, nparts, (float*)d_out);
}